// ALSConvLayer_21019569947059
// MI455X (gfx1250) — compile-verified
//
#include <hip/hip_runtime.h>

typedef __attribute__((ext_vector_type(2))) float v2f;
typedef __attribute__((ext_vector_type(8))) float v8f;

#define ALPHA_F   0.1f
#define ONE_MA_F  0.9f
#define K_ITERS   8

// ---------- atomic helpers ----------
__device__ __forceinline__ void atomic_max_f32(float* addr, float val) {
    // valid ordering trick when the slot is initialized to -inf
    if (val >= 0.0f) atomicMax((int*)addr, __float_as_int(val));
    else             atomicMin((unsigned int*)addr, __float_as_uint(val));
}
__device__ __forceinline__ void atomic_add_f32(float* addr, float val) {
    unsafeAtomicAdd(addr, val);   // lowers to global_atomic_add_f32
}

// =====================================================================
// Y[M,64] = X[M,64] @ W[64,64], fp32, via v_wmma_f32_16x16x4_f32.
// One wave32 handles a 16-row tile; 8 waves per 256-thread block.
// W is staged transposed into LDS (pad 65 -> conflict-free ds loads).
// =====================================================================
__global__ void __launch_bounds__(256)
gemm64_kernel(const float* __restrict__ X, const float* __restrict__ W,
              float* __restrict__ Y, int M)
{
    __shared__ float Wt[64 * 65];   // Wt[n*65 + k] = W[k*64 + n]
    for (int i = threadIdx.x; i < 64 * 64; i += 256) {
        int kk = i >> 6, nn = i & 63;
        Wt[nn * 65 + kk] = W[i];
    }
    __syncthreads();

    const int wave = threadIdx.x >> 5;
    const int lane = threadIdx.x & 31;
    const int half = lane >> 4;     // selects K={0,1} vs K={2,3} within a 4-slice
    const int l16  = lane & 15;
    const int tile = blockIdx.x * 8 + wave;
    const int row0 = tile * 16;
    if (row0 >= M) return;          // uniform per wave

    // A fragments per ISA layout: a[s].x = X[m][4s+2*half], a[s].y = +1
    int r = row0 + l16;
    if (r >= M) r = M - 1;
    const float* xr = X + (size_t)r * 64;
    v2f a[16];
#pragma unroll
    for (int s = 0; s < 16; ++s) {
        const float2 t = *(const float2*)(xr + 4 * s + 2 * half);
        a[s].x = t.x; a[s].y = t.y;
    }

#pragma unroll
    for (int nt = 0; nt < 4; ++nt) {
        const float* wc = &Wt[(nt * 16 + l16) * 65];
        v8f acc = {};
#pragma unroll
        for (int s = 0; s < 16; ++s) {
            v2f b;
            b.x = wc[4 * s + 2 * half];
            b.y = wc[4 * s + 2 * half + 1];
            acc = __builtin_amdgcn_wmma_f32_16x16x4_f32(
                      false, a[s], false, b, (short)0, acc, false, false);
        }
#pragma unroll
        for (int rr = 0; rr < 8; ++rr) {
            int row = row0 + rr + 8 * half;    // D layout: VGPR rr -> M = rr + 8*half
            if (row < M) Y[(size_t)row * 64 + nt * 16 + l16] = acc[rr];
        }
    }
}

// =====================================================================
// softmax state init: m = -inf, denom = 0
// =====================================================================
__global__ void init_softmax_kernel(float* __restrict__ m, float* __restrict__ denom, int n)
{
    int t = blockIdx.x * blockDim.x + threadIdx.x;
    if (t < n) { m[t] = __int_as_float(0xff800000); denom[t] = 0.0f; }
}

// =====================================================================
// per-edge logits + segment max (atomic)
// logit[e,h] = sum_d q[dst][h,d]*(k[src][h,d]+e[e,h,d]) / 4
// =====================================================================
__global__ void logits_kernel(const int* __restrict__ srcI, const int* __restrict__ dstI,
                              const float* __restrict__ q, const float* __restrict__ k,
                              const float* __restrict__ e,
                              float* __restrict__ logits, float* __restrict__ mbuf, int E)
{
    int t = blockIdx.x * blockDim.x + threadIdx.x;
    if (t >= E) return;
    const int s = srcI[t], d = dstI[t];
    const float4* qd = (const float4*)(q + (size_t)d * 64);
    const float4* ks = (const float4*)(k + (size_t)s * 64);
    const float4* ee = (const float4*)(e + (size_t)t * 64);
#pragma unroll
    for (int h = 0; h < 4; ++h) {
        float acc = 0.0f;
#pragma unroll
        for (int j = 0; j < 4; ++j) {
            float4 qv = qd[h * 4 + j];
            float4 kv = ks[h * 4 + j];
            float4 ev = ee[h * 4 + j];
            acc += qv.x * (kv.x + ev.x) + qv.y * (kv.y + ev.y)
                 + qv.z * (kv.z + ev.z) + qv.w * (kv.w + ev.w);
        }
        float logit = acc * 0.25f;              // 1/sqrt(16)
        logits[(size_t)t * 4 + h] = logit;
        atomic_max_f32(&mbuf[(size_t)d * 4 + h], logit);
    }
}

// exp(logit - m[dst]) and segment sum (atomic); ex overwrites logits buffer
__global__ void expsum_kernel(const int* __restrict__ dstI, const float* __restrict__ mbuf,
                              float* __restrict__ attbuf, float* __restrict__ denom, int E)
{
    int t = blockIdx.x * blockDim.x + threadIdx.x;
    if (t >= E) return;
    const int d = dstI[t];
#pragma unroll
    for (int h = 0; h < 4; ++h) {
        float ex = __expf(attbuf[(size_t)t * 4 + h] - mbuf[(size_t)d * 4 + h]);
        attbuf[(size_t)t * 4 + h] = ex;
        atomic_add_f32(&denom[(size_t)d * 4 + h], ex);
    }
}

// att = ex / (denom[dst] + 1e-16), in place
__global__ void attnorm_kernel(const int* __restrict__ dstI, const float* __restrict__ denom,
                               float* __restrict__ attbuf, int E)
{
    int t = blockIdx.x * blockDim.x + threadIdx.x;
    if (t >= E) return;
    const int d = dstI[t];
#pragma unroll
    for (int h = 0; h < 4; ++h)
        attbuf[(size_t)t * 4 + h] /= (denom[(size_t)d * 4 + h] + 1e-16f);
}

// zB = ALPHA * v
__global__ void prop_init_kernel(const float* __restrict__ v, float* __restrict__ zB, int n)
{
    int t = blockIdx.x * blockDim.x + threadIdx.x;
    if (t < n) zB[t] = ALPHA_F * v[t];
}

// zB[dst] += 0.9 * att[e,h] * zA[src]  (per-edge scatter, f32 atomics to L2)
__global__ void scatter_kernel(const int* __restrict__ srcI, const int* __restrict__ dstI,
                               const float* __restrict__ att,
                               const float* __restrict__ zA, float* __restrict__ zB, int E)
{
    int t = blockIdx.x * blockDim.x + threadIdx.x;
    if (t >= E) return;
    const int s = srcI[t], d = dstI[t];
    const float4* za = (const float4*)(zA + (size_t)s * 64);
    float* zb = zB + (size_t)d * 64;
#pragma unroll
    for (int h = 0; h < 4; ++h) {
        const float a = ONE_MA_F * att[(size_t)t * 4 + h];
#pragma unroll
        for (int j = 0; j < 4; ++j) {
            float4 zv = za[h * 4 + j];
            int base = h * 16 + j * 4;
            atomic_add_f32(&zb[base + 0], a * zv.x);
            atomic_add_f32(&zb[base + 1], a * zv.y);
            atomic_add_f32(&zb[base + 2], a * zv.z);
            atomic_add_f32(&zb[base + 3], a * zv.w);
        }
    }
}

// out = x + relu(z)
__global__ void final_kernel(const float* __restrict__ x, const float* __restrict__ z,
                             float* __restrict__ out, int n)
{
    int t = blockIdx.x * blockDim.x + threadIdx.x;
    if (t < n) out[t] = x[t] + fmaxf(z[t], 0.0f);
}

// =====================================================================
extern "C" void kernel_launch(void* const* d_in, const int* in_sizes, int n_in,
                              void* d_out, int out_size, void* d_ws, size_t ws_size,
                              hipStream_t stream)
{
    const float* x  = (const float*)d_in[0];
    const int*   ei = (const int*)d_in[1];   // [2,E] int (harness contract)
    const float* ea = (const float*)d_in[2];
    const float* Wq = (const float*)d_in[3];
    const float* Wk = (const float*)d_in[4];
    const float* Wv = (const float*)d_in[5];
    const float* We = (const float*)d_in[6];
    float* out = (float*)d_out;

    const int N = in_sizes[0] / 64;
    const int E = in_sizes[2] / 64;
    const int* srcI = ei;        // edge_index[0]
    const int* dstI = ei + E;    // edge_index[1]

    // workspace layout (floats)
    float* w = (float*)d_ws;
    size_t off = 0;
    float* q     = w + off; off += (size_t)N * 64;
    float* k     = w + off; off += (size_t)N * 64;
    float* v     = w + off; off += (size_t)N * 64;
    float* att   = w + off; off += (size_t)E * 4;
    float* m     = w + off; off += (size_t)N * 4;
    float* denom = w + off; off += (size_t)N * 4;
    float* e     = w + off;                  // E*64 floats; dead after attnorm
    float* z0    = e;                        // ping-pong buffers alias e region
    float* z1    = e + (size_t)N * 64;

    dim3 blk(256);
    const int nodeTiles = (N + 15) / 16;
    const int edgeTiles = (E + 15) / 16;

    // dense projections (WMMA fp32)
    gemm64_kernel<<<(nodeTiles + 7) / 8, blk, 0, stream>>>(x,  Wq, q, N);
    gemm64_kernel<<<(nodeTiles + 7) / 8, blk, 0, stream>>>(x,  Wk, k, N);
    gemm64_kernel<<<(nodeTiles + 7) / 8, blk, 0, stream>>>(x,  Wv, v, N);
    gemm64_kernel<<<(edgeTiles + 7) / 8, blk, 0, stream>>>(ea, We, e, E);

    // attention
    init_softmax_kernel<<<(N * 4 + 255) / 256, blk, 0, stream>>>(m, denom, N * 4);
    logits_kernel<<<(E + 255) / 256, blk, 0, stream>>>(srcI, dstI, q, k, e, att, m, E);
    expsum_kernel<<<(E + 255) / 256, blk, 0, stream>>>(dstI, m, att, denom, E);
    attnorm_kernel<<<(E + 255) / 256, blk, 0, stream>>>(dstI, denom, att, E);

    // implicit propagation: 8 Jacobi iterations, L2-resident ping-pong
    const float* zA = v;
    float* zB = z0;
    const int nElem = N * 64;
    for (int it = 0; it < K_ITERS; ++it) {
        prop_init_kernel<<<(nElem + 255) / 256, blk, 0, stream>>>(v, zB, nElem);
        scatter_kernel<<<(E + 255) / 256, blk, 0, stream>>>(srcI, dstI, att, zA, zB, E);
        zA = zB;
        zB = (zB == z0) ? z1 : z0;
    }

    // residual + relu
    final_kernel<<<(nElem + 255) / 256, blk, 0, stream>>>(x, zA, out, nElem);
}